// RecurrentGRC_37477884625236
// MI455X (gfx1250) — compile-verified
//
#include <hip/hip_runtime.h>
#include <math.h>

typedef __attribute__((ext_vector_type(2))) float v2f;
typedef __attribute__((ext_vector_type(8))) float v8f;

constexpr int Nn  = 64;     // batch
constexpr int Ss  = 512;    // sequence
constexpr int Dd  = 1024;   // model dim
constexpr int HDd = 512;    // half dim
constexpr int CHh = 2048;   // cell hidden
constexpr int LDP = 68;     // LDS pitch (floats): 16B aligned, conflict-free

// ---- workspace layout (float offsets) ----
constexpr size_t SZ_STATE   = (size_t)Ss * Nn * HDd;        // 16,777,216 floats
constexpr size_t OFF_FSTATE = 0;
constexpr size_t OFF_BSTATE = OFF_FSTATE + SZ_STATE;
constexpr size_t OFF_FHS    = OFF_BSTATE + SZ_STATE;
constexpr size_t OFF_BHS    = OFF_FHS + SZ_STATE;
constexpr size_t OFF_INTERF = OFF_BHS + SZ_STATE;
constexpr size_t OFF_INTERB = OFF_INTERF + (size_t)Nn * CHh;
constexpr size_t OFF_UF     = OFF_INTERB + (size_t)Nn * CHh;
constexpr size_t OFF_UB     = OFF_UF + (size_t)Nn * HDd;
constexpr size_t OFF_HF     = OFF_UB + (size_t)Nn * HDd;
constexpr size_t OFF_HB     = OFF_HF + (size_t)Nn * HDd;
constexpr size_t OFF_BAR    = OFF_HB + (size_t)Nn * HDd;    // unsigned[64]

// ---------------------------------------------------------------- helpers
__device__ __forceinline__ float gelu_exact(float x) {
    return 0.5f * x * (1.0f + erff(x * 0.70710678118654752f));
}
__device__ __forceinline__ float sigm(float x) {
    return 1.0f / (1.0f + expf(-x));
}
__device__ __forceinline__ v8f bias_fill(const float* b, int col) {
    float bv = b[col];
    v8f c;
#pragma unroll
    for (int i = 0; i < 8; ++i) c[i] = bv;
    return c;
}

// 64x64 output tile GEMM-accumulate using V_WMMA_F32_16X16X4_F32.
// Block = 256 threads = 8 waves; wave w owns two 16x16 tiles:
//   m-tile = w&3, n-tiles = (w>>2)*2 and (w>>2)*2+1 (virtual cols 0..63).
// B columns are taken as 4 groups of 16 contiguous columns starting at bc[g]
// (virtual col v -> actual col bc[v>>4] + (v&15)); identity = {c0,c0+16,..}.
// A (64 x K, row-major lda) and B (K x *, row-major ldb) are staged in LDS
// in 64-wide K chunks.
__device__ __forceinline__ void gemm64_acc(
    v8f& c0, v8f& c1,
    const float* __restrict__ A, int lda,
    const float* __restrict__ B, int ldb,
    int K, const int* bc,
    float* ldsA, float* ldsB)
{
    const int tid   = threadIdx.x;
    const int lane  = tid & 31;
    const int wave  = tid >> 5;
    const int mt    = wave & 3;
    const int np    = wave >> 2;
    const int lrow  = tid >> 2;            // 0..63 (staging row)
    const int lq    = tid & 3;             // 0..3  (16-col group)
    const int mrow  = mt * 16 + (lane & 15);
    const int nc    = lane & 15;
    const int khalf = (lane >> 4) << 1;    // 0 or 2
    const int v0    = np * 32 + nc;        // virtual col of tile 0
    const int v1    = v0 + 16;             // virtual col of tile 1

    for (int kc = 0; kc < K; kc += 64) {
        __syncthreads();
        {   // stage A chunk: 64 rows x 64 k
            const float* src = A + (size_t)lrow * lda + kc + lq * 16;
            float* dst = ldsA + lrow * LDP + lq * 16;
#pragma unroll
            for (int i = 0; i < 4; ++i)
                *reinterpret_cast<float4*>(dst + i * 4) =
                    *reinterpret_cast<const float4*>(src + i * 4);
        }
        {   // stage B chunk: 64 k-rows x 4 groups of 16 cols
            const float* src = B + (size_t)(kc + lrow) * ldb + bc[lq];
            float* dst = ldsB + lrow * LDP + lq * 16;
#pragma unroll
            for (int i = 0; i < 4; ++i)
                *reinterpret_cast<float4*>(dst + i * 4) =
                    *reinterpret_cast<const float4*>(src + i * 4);
        }
        __syncthreads();
#pragma unroll
        for (int k = 0; k < 64; k += 4) {
            const int kk = k + khalf;
            v2f a, b0, b1;
            a.x  = ldsA[mrow * LDP + kk];
            a.y  = ldsA[mrow * LDP + kk + 1];
            b0.x = ldsB[kk * LDP + v0];
            b0.y = ldsB[(kk + 1) * LDP + v0];
            b1.x = ldsB[kk * LDP + v1];
            b1.y = ldsB[(kk + 1) * LDP + v1];
            c0 = __builtin_amdgcn_wmma_f32_16x16x4_f32(
                     false, a, false, b0, (short)0, c0, false, false);
            c1 = __builtin_amdgcn_wmma_f32_16x16x4_f32(
                     false, a, false, b1, (short)0, c1, false, false);
        }
    }
}

// C/D fragment layout: VGPR i -> rows (mt*16 + (lane>=16?8:0) + i), col = lane&15.
__device__ __forceinline__ void store_tile(const v8f& c, float* out, int ldo,
                                           int row0, int col) {
#pragma unroll
    for (int i = 0; i < 8; ++i) out[(size_t)(row0 + i) * ldo + col] = c[i];
}

// agent-scope monotonic grid barrier among 32 blocks sharing one counter
__device__ __forceinline__ void grid_barrier(unsigned* cnt, unsigned target) {
    __syncthreads();
    if (threadIdx.x == 0) {
        __threadfence();
        __hip_atomic_fetch_add(cnt, 1u, __ATOMIC_RELEASE, __HIP_MEMORY_SCOPE_AGENT);
        while (__hip_atomic_load(cnt, __ATOMIC_ACQUIRE, __HIP_MEMORY_SCOPE_AGENT) < target)
            __builtin_amdgcn_s_sleep(1);
    }
    __syncthreads();
}

// ---------------------------------------------------------------- kernels

// zero recurrent states + barrier counters
__global__ void init_kernel(float* hf, float* hb, unsigned* bar) {
    int i = blockIdx.x * 256 + threadIdx.x;
    if (i < Nn * HDd) { hf[i] = 0.0f; hb[i] = 0.0f; }
    if (i < 64) bar[i] = 0u;
}

// state_pre = x @ it_w + it_b, written split into fstate/bstate [S][N][HD]
__global__ __launch_bounds__(256) void input_gemm_kernel(
    const float* __restrict__ x, const float* __restrict__ it_w,
    const float* __restrict__ it_b,
    float* __restrict__ fstate, float* __restrict__ bstate)
{
    __shared__ float ldsA[64 * LDP];
    __shared__ float ldsB[64 * LDP];
    const int cb = blockIdx.x;           // 0..15  (64 cols each)
    const int rb = blockIdx.y;           // 0..511 (64 rows each)
    const int r0 = rb * 64;
    const int n  = r0 >> 9;              // batch index (S=512 rows per n)
    const int s0 = r0 & 511;
    const int c0col = cb * 64;
    const int lane = threadIdx.x & 31;
    const int wave = threadIdx.x >> 5;
    const int mt = wave & 3, np = wave >> 2;
    const int nc = lane & 15;
    const int rhalf = (lane >> 4) << 3;

    const int colA = c0col + np * 32 + nc;
    const int colB = colA + 16;
    v8f c0 = bias_fill(it_b, colA);
    v8f c1 = bias_fill(it_b, colB);
    int bc[4] = {c0col, c0col + 16, c0col + 32, c0col + 48};
    gemm64_acc(c0, c1, x + (size_t)r0 * Dd, Dd, it_w, Dd, Dd, bc, ldsA, ldsB);

    // scatter into [S][N][HD] split layout
    float* outA = (colA < HDd) ? fstate : bstate;
    float* outB = (colB < HDd) ? fstate : bstate;
    const int ca = colA & (HDd - 1);
    const int cbb = colB & (HDd - 1);
    const int row0 = mt * 16 + rhalf;
#pragma unroll
    for (int i = 0; i < 8; ++i) {
        const size_t sr = (size_t)(s0 + row0 + i) * Nn + n;
        outA[sr * HDd + ca]  = c0[i];
        outB[sr * HDd + cbb] = c1[i];
    }
}

// in-place LayerNorm over the (split) 1024-wide rows, scale it_g/it_beta
__global__ __launch_bounds__(256) void ln_rows_kernel(
    float* __restrict__ fstate, float* __restrict__ bstate,
    const float* __restrict__ g, const float* __restrict__ beta)
{
    __shared__ float r1[256], r2[256];
    const int r = blockIdx.x;            // 0..N*S-1
    const int n = r >> 9, s = r & 511;
    const size_t base = ((size_t)s * Nn + n) * HDd;
    const int tid = threadIdx.x;
    float v0 = fstate[base + tid];
    float v1 = fstate[base + tid + 256];
    float v2 = bstate[base + tid];
    float v3 = bstate[base + tid + 256];
    r1[tid] = v0 + v1 + v2 + v3;
    r2[tid] = v0 * v0 + v1 * v1 + v2 * v2 + v3 * v3;
    __syncthreads();
    for (int off = 128; off > 0; off >>= 1) {
        if (tid < off) { r1[tid] += r1[tid + off]; r2[tid] += r2[tid + off]; }
        __syncthreads();
    }
    const float mu = r1[0] * (1.0f / Dd);
    const float var = r2[0] * (1.0f / Dd) - mu * mu;
    const float rs = rsqrtf(var + 1e-5f);
    fstate[base + tid]        = (v0 - mu) * rs * g[tid]        + beta[tid];
    fstate[base + tid + 256]  = (v1 - mu) * rs * g[tid + 256]  + beta[tid + 256];
    bstate[base + tid]        = (v2 - mu) * rs * g[tid + 512]  + beta[tid + 512];
    bstate[base + tid + 256]  = (v3 - mu) * rs * g[tid + 768]  + beta[tid + 768];
}

// persistent bidirectional masked GRC scan: 64 blocks (32 fwd + 32 bwd)
__global__ __launch_bounds__(256) void scan_kernel(
    const float* __restrict__ fstate, const float* __restrict__ bstate,
    const float* __restrict__ mask,
    const float* __restrict__ fw1, const float* __restrict__ fb1,
    const float* __restrict__ fw2, const float* __restrict__ fb2,
    const float* __restrict__ fg,  const float* __restrict__ fbeta,
    const float* __restrict__ bw1, const float* __restrict__ bb1,
    const float* __restrict__ bw2, const float* __restrict__ bb2,
    const float* __restrict__ bg,  const float* __restrict__ bbeta,
    float* __restrict__ f_hs, float* __restrict__ b_hs,
    float* __restrict__ inter_f, float* __restrict__ inter_b,
    float* __restrict__ u_f, float* __restrict__ u_b,
    float* __restrict__ h_f, float* __restrict__ h_b,
    unsigned* __restrict__ barv)
{
    __shared__ float ldsA[64 * LDP];
    __shared__ float ldsB[64 * LDP];

    const int dir = blockIdx.x >> 5;     // 0 = forward, 1 = backward
    const int blk = blockIdx.x & 31;
    const float* xst  = dir ? bstate : fstate;
    const float* w1   = dir ? bw1 : fw1;
    const float* b1   = dir ? bb1 : fb1;
    const float* w2   = dir ? bw2 : fw2;
    const float* b2   = dir ? bb2 : fb2;
    const float* lng  = dir ? bg : fg;
    const float* lnb  = dir ? bbeta : fbeta;
    float* inter = dir ? inter_b : inter_f;
    float* u     = dir ? u_b : u_f;
    float* h     = dir ? h_b : h_f;
    float* hist  = dir ? b_hs : f_hs;
    unsigned* cnt = barv + dir * 16;     // separate cachelines per direction

    const int tid = threadIdx.x;
    const int lane = tid & 31;
    const int wave = tid >> 5;
    const int mt = wave & 3, np = wave >> 2;
    const int nc = lane & 15;
    const int rhalf = (lane >> 4) << 3;
    unsigned phase = 0;

    for (int step = 0; step < Ss; ++step) {
        const int t = dir ? (Ss - 1 - step) : step;
        const float* x_t = xst + (size_t)t * Nn * HDd;

        // h from previous step's phase C is visible to everyone
        grid_barrier(cnt, (++phase) * 32u);

        // ---- Phase A: inter[:, blk*64 : blk*64+64] = gelu([h, x_t] @ w1 + b1)
        {
            const int ncol0 = blk * 64;
            const int colA = ncol0 + np * 32 + nc;
            const int colB = colA + 16;
            v8f c0 = bias_fill(b1, colA);
            v8f c1 = bias_fill(b1, colB);
            int bc[4] = {ncol0, ncol0 + 16, ncol0 + 32, ncol0 + 48};
            gemm64_acc(c0, c1, h, HDd, w1, CHh, HDd, bc, ldsA, ldsB);
            gemm64_acc(c0, c1, x_t, HDd, w1 + (size_t)HDd * CHh, CHh, HDd, bc,
                       ldsA, ldsB);
#pragma unroll
            for (int i = 0; i < 8; ++i) { c0[i] = gelu_exact(c0[i]); c1[i] = gelu_exact(c1[i]); }
            store_tile(c0, inter, CHh, mt * 16 + rhalf, colA);
            store_tile(c1, inter, CHh, mt * 16 + rhalf, colB);
        }
        grid_barrier(cnt, (++phase) * 32u);

        // ---- Phase B: gate columns j0..j0+15 of contents = inter @ w2 + b2,
        //       then u = sig(g0)*h + sig(g1)*x + sig(g2)*g3
        {
            const int j0 = blk * 16;
            int bc2[4] = {j0, HDd + j0, 2 * HDd + j0, 3 * HDd + j0};
            v8f c0 = bias_fill(b2, bc2[np * 2] + nc);
            v8f c1 = bias_fill(b2, bc2[np * 2 + 1] + nc);
            gemm64_acc(c0, c1, inter, CHh, w2, 4 * HDd, CHh, bc2, ldsA, ldsB);
            __syncthreads();   // done reading staged chunks; reuse ldsA
            {   // contents tiles -> LDS, virtual col layout [64][64]
                const int cv0 = np * 32 + nc;
#pragma unroll
                for (int i = 0; i < 8; ++i) {
                    ldsA[(mt * 16 + rhalf + i) * LDP + cv0]      = c0[i];
                    ldsA[(mt * 16 + rhalf + i) * LDP + cv0 + 16] = c1[i];
                }
            }
            __syncthreads();
            for (int e = tid; e < Nn * 16; e += 256) {
                const int m = e >> 4, jj = e & 15;
                const float g0 = ldsA[m * LDP + jj];
                const float g1 = ldsA[m * LDP + 16 + jj];
                const float g2 = ldsA[m * LDP + 32 + jj];
                const float g3 = ldsA[m * LDP + 48 + jj];
                const int col = j0 + jj;
                const float uv = sigm(g0) * h[m * HDd + col] +
                                 sigm(g1) * x_t[(size_t)m * HDd + col] +
                                 sigm(g2) * g3;
                u[m * HDd + col] = uv;
            }
        }
        grid_barrier(cnt, (++phase) * 32u);

        // ---- Phase C: LayerNorm rows 2*blk, 2*blk+1; mask blend; record
        {
            float* red1 = ldsA;
            float* red2 = ldsA + 256;
            for (int rr = 0; rr < 2; ++rr) {
                const int n = blk * 2 + rr;
                const float* urow = u + n * HDd;
                const float v0 = urow[tid], v1 = urow[tid + 256];
                red1[tid] = v0 + v1;
                red2[tid] = v0 * v0 + v1 * v1;
                __syncthreads();
                for (int off = 128; off > 0; off >>= 1) {
                    if (tid < off) { red1[tid] += red1[tid + off]; red2[tid] += red2[tid + off]; }
                    __syncthreads();
                }
                const float mu  = red1[0] * (1.0f / HDd);
                const float var = red2[0] * (1.0f / HDd) - mu * mu;
                const float rs  = rsqrtf(var + 1e-5f);
                const float mv  = mask[(size_t)n * Ss + t];
#pragma unroll
                for (int q = 0; q < 2; ++q) {
                    const int j = tid + q * 256;
                    const float hn = ((q ? v1 : v0) - mu) * rs * lng[j] + lnb[j];
                    const float hold = h[n * HDd + j];
                    const float hv = mv * hn + (1.0f - mv) * hold;
                    h[n * HDd + j] = hv;
                    hist[((size_t)t * Nn + n) * HDd + j] = hv;
                }
                __syncthreads();
            }
        }
    }
}

// seq = concat(f_hs, b_hs) @ out_w + out_b  -> d_out[(n*S+s)*D + c]
__global__ __launch_bounds__(256) void out_gemm_kernel(
    const float* __restrict__ f_hs, const float* __restrict__ b_hs,
    const float* __restrict__ out_w, const float* __restrict__ out_b,
    float* __restrict__ seq)
{
    __shared__ float ldsA[64 * LDP];
    __shared__ float ldsB[64 * LDP];
    const int cb = blockIdx.x;
    const int rb = blockIdx.y;
    const int r0 = rb * 64;
    const int n  = r0 >> 9;
    const int s0 = r0 & 511;
    const int c0col = cb * 64;
    const int lane = threadIdx.x & 31;
    const int wave = threadIdx.x >> 5;
    const int mt = wave & 3, np = wave >> 2;
    const int nc = lane & 15;
    const int rhalf = (lane >> 4) << 3;

    const int colA = c0col + np * 32 + nc;
    const int colB = colA + 16;
    v8f c0 = bias_fill(out_b, colA);
    v8f c1 = bias_fill(out_b, colB);
    int bc[4] = {c0col, c0col + 16, c0col + 32, c0col + 48};
    const size_t abase = ((size_t)s0 * Nn + n) * HDd;
    gemm64_acc(c0, c1, f_hs + abase, Nn * HDd, out_w, Dd, HDd, bc, ldsA, ldsB);
    gemm64_acc(c0, c1, b_hs + abase, Nn * HDd, out_w + (size_t)HDd * Dd, Dd,
               HDd, bc, ldsA, ldsB);
    store_tile(c0, seq + (size_t)r0 * Dd, Dd, mt * 16 + rhalf, colA);
    store_tile(c1, seq + (size_t)r0 * Dd, Dd, mt * 16 + rhalf, colB);
}

// global_state[n][d] = mean over s of seq
__global__ __launch_bounds__(256) void gmean_kernel(
    const float* __restrict__ seq, float* __restrict__ gs)
{
    const int e = blockIdx.x * 256 + threadIdx.x;   // 0..N*D-1
    const int n = e >> 10, d = e & 1023;
    float acc = 0.0f;
    const float* p = seq + ((size_t)n * Ss) * Dd + d;
    for (int s = 0; s < Ss; ++s) acc += p[(size_t)s * Dd];
    gs[e] = acc * (1.0f / Ss);
}

// ---------------------------------------------------------------- launch
extern "C" void kernel_launch(void* const* d_in, const int* in_sizes, int n_in,
                              void* d_out, int out_size, void* d_ws, size_t ws_size,
                              hipStream_t stream) {
    const float* x       = (const float*)d_in[0];
    const float* mask    = (const float*)d_in[1];
    const float* it_w    = (const float*)d_in[2];
    const float* it_b    = (const float*)d_in[3];
    const float* it_g    = (const float*)d_in[4];
    const float* it_beta = (const float*)d_in[5];
    const float* fw1     = (const float*)d_in[6];
    const float* fb1     = (const float*)d_in[7];
    const float* fw2     = (const float*)d_in[8];
    const float* fb2     = (const float*)d_in[9];
    const float* f_ln_g  = (const float*)d_in[10];
    const float* f_ln_b  = (const float*)d_in[11];
    const float* bw1     = (const float*)d_in[12];
    const float* bb1     = (const float*)d_in[13];
    const float* bw2     = (const float*)d_in[14];
    const float* bb2     = (const float*)d_in[15];
    const float* b_ln_g  = (const float*)d_in[16];
    const float* b_ln_b  = (const float*)d_in[17];
    const float* out_w   = (const float*)d_in[18];
    const float* out_b   = (const float*)d_in[19];

    float* ws = (float*)d_ws;
    float* fstate  = ws + OFF_FSTATE;
    float* bstate  = ws + OFF_BSTATE;
    float* f_hs    = ws + OFF_FHS;
    float* b_hs    = ws + OFF_BHS;
    float* inter_f = ws + OFF_INTERF;
    float* inter_b = ws + OFF_INTERB;
    float* u_f     = ws + OFF_UF;
    float* u_b     = ws + OFF_UB;
    float* h_f     = ws + OFF_HF;
    float* h_b     = ws + OFF_HB;
    unsigned* barv = (unsigned*)(ws + OFF_BAR);

    float* seq = (float*)d_out;
    float* gs  = seq + (size_t)Nn * Ss * Dd;

    init_kernel<<<dim3((Nn * HDd + 255) / 256), dim3(256), 0, stream>>>(h_f, h_b, barv);
    input_gemm_kernel<<<dim3(Dd / 64, (Nn * Ss) / 64), dim3(256), 0, stream>>>(
        x, it_w, it_b, fstate, bstate);
    ln_rows_kernel<<<dim3(Nn * Ss), dim3(256), 0, stream>>>(fstate, bstate, it_g, it_beta);
    scan_kernel<<<dim3(64), dim3(256), 0, stream>>>(
        fstate, bstate, mask,
        fw1, fb1, fw2, fb2, f_ln_g, f_ln_b,
        bw1, bb1, bw2, bb2, b_ln_g, b_ln_b,
        f_hs, b_hs, inter_f, inter_b, u_f, u_b, h_f, h_b, barv);
    out_gemm_kernel<<<dim3(Dd / 64, (Nn * Ss) / 64), dim3(256), 0, stream>>>(
        f_hs, b_hs, out_w, out_b, seq);
    gmean_kernel<<<dim3((Nn * Dd) / 256), dim3(256), 0, stream>>>(seq, gs);
}